// PolyConv_15453292331333
// MI455X (gfx1250) — compile-verified
//
#include <hip/hip_runtime.h>
#include <hip/hip_bf16.h>

typedef float v2f __attribute__((ext_vector_type(2)));
typedef float v8f __attribute__((ext_vector_type(8)));

#define FD 128          // feature dim
#define AST 132         // padded LDS stride (conflict-free: bank = 4*mn (+2))

// ---------------- utility kernels ----------------

__global__ void pc_zero_f32(float* __restrict__ p, int n) {
    int i = blockIdx.x * blockDim.x + threadIdx.x;
    if (i < n) p[i] = 0.0f;
}

__global__ void pc_degree(const int* __restrict__ dst, float* __restrict__ deg, int n_edges) {
    int e = blockIdx.x * blockDim.x + threadIdx.x;
    if (e < n_edges) atomicAdd(&deg[dst[e]], 1.0f);
}

__global__ void pc_dinv(const float* __restrict__ deg, float* __restrict__ dinv, int n) {
    int i = blockIdx.x * blockDim.x + threadIdx.x;
    if (i < n) {
        float d = deg[i];
        d = d < 1.0f ? 1.0f : d;
        dinv[i] = 1.0f / sqrtf(d);
    }
}

// ---------------- WMMA GEMM: h = (0.5*feat*diag0) @ W^T + b ----------------
// One block = 16 output rows (nodes). 8 waves = 8 column tiles of 16.
// A (16x4 f32):   lane%16 = M, K = vgpr + 2*(lane/16)
// B (4x16 f32):   lane%16 = N, K = vgpr + 2*(lane/16)   (B[k][n] = W[n][k])
// C/D (16x16 f32): lane%16 = N, M = r + 8*(lane/16)
__global__ void pc_gemm_wmma(const float* __restrict__ feat,
                             const float* __restrict__ diag0,
                             const float* __restrict__ W,
                             const float* __restrict__ bias,
                             float* __restrict__ h, int n_nodes) {
    __shared__ float As[16 * AST];

    const int row0 = blockIdx.x * 16;
    const int tid  = threadIdx.x;

    // Stage pre-scaled A tile: As[m][k] = 0.5 * feat[row0+m][k] * diag0[k]
    for (int i = tid; i < 16 * FD; i += 256) {
        int m = i >> 7;
        int k = i & (FD - 1);
        int r = row0 + m;
        float v = 0.0f;
        if (r < n_nodes) v = 0.5f * feat[(size_t)r * FD + k] * diag0[k];
        As[m * AST + k] = v;
    }
    __syncthreads();

    const int wave = tid >> 5;        // column tile 0..7
    const int lane = tid & 31;
    const int g    = lane >> 4;       // half-wave 0/1
    const int mn   = lane & 15;       // M for A, N for B/C/D
    const int col0 = wave * 16;

    const float* Wrow = W + (size_t)(col0 + mn) * FD;  // B[k][n] = W[n][k]
    const float* Arow = As + mn * AST;

    v8f acc = {};
#pragma unroll
    for (int k0 = 0; k0 < FD; k0 += 4) {
        const int kk = k0 + 2 * g;
        v2f a, b;
        a.x = Arow[kk];
        a.y = Arow[kk + 1];
        b.x = Wrow[kk];
        b.y = Wrow[kk + 1];
        acc = __builtin_amdgcn_wmma_f32_16x16x4_f32(
            /*neg_a=*/false, a, /*neg_b=*/false, b,
            /*c_mod=*/(short)0, acc, /*reuse_a=*/false, /*reuse_b=*/false);
    }

    const float bb = bias[col0 + mn];
#pragma unroll
    for (int r = 0; r < 8; ++r) {
        int m = r + 8 * g;
        int row = row0 + m;
        if (row < n_nodes) h[(size_t)row * FD + col0 + mn] = acc[r] + bb;
    }
}

// ---------------- edge scatter: agg[dst] += feat_cur[src] * dinv[src] ----------------
// 32 threads per edge, float4 per thread (32*4 = 128 features).
__global__ void pc_scatter(const int* __restrict__ src, const int* __restrict__ dst,
                           const float* __restrict__ feat_cur,
                           const float* __restrict__ dinv,
                           float* __restrict__ agg, int n_edges) {
    int t = blockIdx.x * blockDim.x + threadIdx.x;
    int e = t >> 5;
    if (e >= n_edges) return;
    int c = (t & 31) * 4;
    int s = src[e];
    int d = dst[e];
    float sc = dinv[s];
    const float4 v = *(const float4*)(feat_cur + (size_t)s * FD + c);
    float* out = agg + (size_t)d * FD + c;
    atomicAdd(out + 0, v.x * sc);
    atomicAdd(out + 1, v.y * sc);
    atomicAdd(out + 2, v.z * sc);
    atomicAdd(out + 3, v.w * sc);
}

// ---------------- node update: feat -= agg*dinv*diag_k ; h += theta_k*feat ----------------
__global__ void pc_update(float* __restrict__ feat_cur, const float* __restrict__ agg,
                          const float* __restrict__ dinv, const float* __restrict__ diag_k,
                          float theta_k, float* __restrict__ h, int total) {
    int i = blockIdx.x * blockDim.x + threadIdx.x;
    if (i >= total) return;
    int node = i >> 7;
    int k    = i & (FD - 1);
    float f = feat_cur[i] - agg[i] * dinv[node] * diag_k[k];
    feat_cur[i] = f;
    h[i] += theta_k * f;
}

// ---------------- launcher ----------------

extern "C" void kernel_launch(void* const* d_in, const int* in_sizes, int n_in,
                              void* d_out, int out_size, void* d_ws, size_t ws_size,
                              hipStream_t stream) {
    const float* feat = (const float*)d_in[0];
    const int*   ei   = (const int*)d_in[1];   // (2, E): row 0 = src, row 1 = dst
    const float* diag = (const float*)d_in[2]; // (3, 128)
    const float* W    = (const float*)d_in[3];
    const float* bias = (const float*)d_in[4];
    float* h = (float*)d_out;

    const int N = in_sizes[0] / FD;
    const int E = in_sizes[1] / 2;
    const int* src = ei;
    const int* dst = ei + E;

    // workspace carve-up
    float* feat_cur = (float*)d_ws;
    float* agg      = feat_cur + (size_t)N * FD;
    float* deg      = agg + (size_t)N * FD;
    float* dinv     = deg + N;

    const int total = N * FD;
    const float THETA[3] = {0.5f, -1.0f, 0.5f};

    // degree + d^{-1/2}
    pc_zero_f32<<<(N + 255) / 256, 256, 0, stream>>>(deg, N);
    pc_degree<<<(E + 255) / 256, 256, 0, stream>>>(dst, deg, E);
    pc_dinv<<<(N + 255) / 256, 256, 0, stream>>>(deg, dinv, N);

    // h = (0.5*feat*diag0) @ W^T + b   (WMMA)
    pc_gemm_wmma<<<(N + 15) / 16, 256, 0, stream>>>(feat, diag, W, bias, h, N);

    // feat_cur = feat (do not mutate inputs)
    hipMemcpyAsync(feat_cur, feat, (size_t)total * sizeof(float),
                   hipMemcpyDeviceToDevice, stream);

    for (int k = 1; k <= 2; ++k) {
        pc_zero_f32<<<(total + 255) / 256, 256, 0, stream>>>(agg, total);
        int sth = E * 32;
        pc_scatter<<<(sth + 255) / 256, 256, 0, stream>>>(src, dst, feat_cur, dinv, agg, E);
        pc_update<<<(total + 255) / 256, 256, 0, stream>>>(feat_cur, agg, dinv,
                                                           diag + (size_t)k * FD,
                                                           THETA[k], h, total);
    }
}